// RasterizeGaussians_5420248727854
// MI455X (gfx1250) — compile-verified
//
#include <hip/hip_runtime.h>
#include <cstdint>

// ---------------------------------------------------------------------------
// Gaussian splat rasterizer for gfx1250 (MI455X).
//  P=8192 gaussians, 256x256 image, 3 channels.
//  k1: per-gaussian preprocess -> depth + 12-float record (3 x float4)
//  k2: single-block bitonic sort by (depth, idx) in LDS + gather sorted recs
//  k3: rasterize: per wave 16 pixels; alpha/transmittance in f32 VALU,
//      color accumulation via V_WMMA_F32_16X16X4_F32 (exact f32),
//      gaussian chunks staged LDS via global_load_async_to_lds_b128
//      (double buffered, s_wait_asynccnt).
// ---------------------------------------------------------------------------

#define GP    8192
#define GW    256
#define GH    256
#define GFX   256.0f
#define GFY   256.0f
#define GTANX 0.5f
#define GTANY 0.5f

#define NG_CHUNK   256                   // gaussians staged per chunk
#define NCHUNK     (GP / NG_CHUNK)       // 32
#define QUADS      (NG_CHUNK * 3)        // float4 per chunk = 768 (12 KB)

typedef __attribute__((ext_vector_type(2))) float v2f;
typedef __attribute__((ext_vector_type(8))) float v8f;

// ---------------------------------------------------------------------------
// Kernel 1: preprocess
// rec layout (3 float4 per gaussian):
//   r0 = (mean2.x, mean2.y, conic0, conic1)
//   r1 = (conic2, opacity, col0, col1)
//   r2 = (col2, 0, 0, 0)
// ---------------------------------------------------------------------------
__global__ __launch_bounds__(256) void gs_preprocess(
    const float* __restrict__ means, const float* __restrict__ colors,
    const float* __restrict__ opac,  const float* __restrict__ scales,
    const float* __restrict__ rots,  const float* __restrict__ V,
    const float* __restrict__ Pm,
    float* __restrict__ depths, float4* __restrict__ recs) {
  int p = blockIdx.x * 256 + threadIdx.x;
  if (p >= GP) return;

  float qr = rots[4*p+0], qx = rots[4*p+1], qy = rots[4*p+2], qz = rots[4*p+3];
  float qn = 1.0f / sqrtf(qr*qr + qx*qx + qy*qy + qz*qz);
  qr *= qn; qx *= qn; qy *= qn; qz *= qn;

  float R[3][3];
  R[0][0] = 1.f - 2.f*(qy*qy + qz*qz); R[0][1] = 2.f*(qx*qy - qr*qz); R[0][2] = 2.f*(qx*qz + qr*qy);
  R[1][0] = 2.f*(qx*qy + qr*qz); R[1][1] = 1.f - 2.f*(qx*qx + qz*qz); R[1][2] = 2.f*(qy*qz - qr*qx);
  R[2][0] = 2.f*(qx*qz - qr*qy); R[2][1] = 2.f*(qy*qz + qr*qx); R[2][2] = 1.f - 2.f*(qx*qx + qy*qy);

  float s0 = scales[3*p], s1 = scales[3*p+1], s2v = scales[3*p+2];
  float s2[3] = { s0*s0, s1*s1, s2v*s2v };

  float Sg[3][3];
#pragma unroll
  for (int i = 0; i < 3; ++i)
#pragma unroll
    for (int k = 0; k < 3; ++k)
      Sg[i][k] = R[i][0]*s2[0]*R[k][0] + R[i][1]*s2[1]*R[k][1] + R[i][2]*s2[2]*R[k][2];

  float mx = means[3*p], my = means[3*p+1], mz = means[3*p+2];

  float pv[3];
#pragma unroll
  for (int j = 0; j < 3; ++j)
    pv[j] = mx*V[0*4+j] + my*V[1*4+j] + mz*V[2*4+j] + V[3*4+j];

  float ph[4];
#pragma unroll
  for (int j = 0; j < 4; ++j)
    ph[j] = mx*Pm[0*4+j] + my*Pm[1*4+j] + mz*Pm[2*4+j] + Pm[3*4+j];

  float ipw = 1.0f / (ph[3] + 1e-7f);
  float ppx = ph[0]*ipw, ppy = ph[1]*ipw;
  float m2x = ((ppx + 1.0f) * (float)GW - 1.0f) * 0.5f;
  float m2y = ((ppy + 1.0f) * (float)GH - 1.0f) * 0.5f;

  // cov_cam = Rv^T * Sigma * Rv  (Rv = V[:3,:3])
  float M1[3][3];
#pragma unroll
  for (int j = 0; j < 3; ++j)
#pragma unroll
    for (int l = 0; l < 3; ++l)
      M1[j][l] = Sg[j][0]*V[0*4+l] + Sg[j][1]*V[1*4+l] + Sg[j][2]*V[2*4+l];
  float C[3][3];
#pragma unroll
  for (int i = 0; i < 3; ++i)
#pragma unroll
    for (int l = 0; l < 3; ++l)
      C[i][l] = V[0*4+i]*M1[0][l] + V[1*4+i]*M1[1][l] + V[2*4+i]*M1[2][l];

  float tz = pv[2];
  float iz = 1.0f / tz;
  float tx = fminf(fmaxf(pv[0]*iz, -1.3f*GTANX), 1.3f*GTANX) * tz;
  float ty = fminf(fmaxf(pv[1]*iz, -1.3f*GTANY), 1.3f*GTANY) * tz;
  float J00 = GFX*iz, J02 = -GFX*tx*iz*iz;
  float J11 = GFY*iz, J12 = -GFY*ty*iz*iz;

  // cov2d = J * C * J^T
  float u0 = J00*C[0][0] + J02*C[2][0];
  float u1 = J00*C[0][1] + J02*C[2][1];
  float u2 = J00*C[0][2] + J02*C[2][2];
  float v1 = J11*C[1][1] + J12*C[2][1];
  float v2 = J11*C[1][2] + J12*C[2][2];
  float c00 = u0*J00 + u2*J02;
  float c01 = u1*J11 + u2*J12;
  float c11 = v1*J11 + v2*J12;

  float a  = c00 + 0.3f;
  float b  = c01;
  float cc = c11 + 0.3f;
  float idet = 1.0f / (a*cc - b*b);
  float cn0 = cc*idet, cn1 = -b*idet, cn2 = a*idet;

  depths[p] = pv[2];
  recs[3*p+0] = make_float4(m2x, m2y, cn0, cn1);
  recs[3*p+1] = make_float4(cn2, opac[p], colors[3*p+0], colors[3*p+1]);
  recs[3*p+2] = make_float4(colors[3*p+2], 0.f, 0.f, 0.f);
}

// ---------------------------------------------------------------------------
// Kernel 2: single-block bitonic sort of 8192 (key, idx) in LDS (64 KB of the
// 320 KB WGP LDS), lexicographic (depth, idx) => equivalent to stable argsort.
// Then gather unsorted records into sorted order.
// ---------------------------------------------------------------------------
__global__ __launch_bounds__(1024) void gs_sort_gather(
    const float* __restrict__ depths, const float4* __restrict__ recs_in,
    float4* __restrict__ recs_out) {
  __shared__ float key[GP];
  __shared__ int   idx[GP];
  const int tid = threadIdx.x;

  for (int i = tid; i < GP; i += 1024) { key[i] = depths[i]; idx[i] = i; }
  __syncthreads();

  for (int k = 2; k <= GP; k <<= 1) {
    for (int j = k >> 1; j > 0; j >>= 1) {
      for (int i = tid; i < GP; i += 1024) {
        int ixj = i ^ j;
        if (ixj > i) {
          bool up = ((i & k) == 0);
          float ka = key[i], kb = key[ixj];
          int   ia = idx[i], ib = idx[ixj];
          bool a_gt_b = (ka > kb) || ((ka == kb) && (ia > ib));
          if (a_gt_b == up) {
            key[i] = kb; key[ixj] = ka;
            idx[i] = ib; idx[ixj] = ia;
          }
        }
      }
      __syncthreads();
    }
  }

  for (int s = tid; s < GP; s += 1024) {
    int g = idx[s];
    recs_out[3*s+0] = recs_in[3*g+0];
    recs_out[3*s+1] = recs_in[3*g+1];
    recs_out[3*s+2] = recs_in[3*g+2];
  }
}

// ---------------------------------------------------------------------------
// Kernel 3: rasterize.
// Wave = 16 pixels. Lanes 0-15: pixel L, gaussians K0,K1 of each 4-group;
// lanes 16-31: pixel L, gaussians K2,K3 (matches f32 16x4 A-matrix layout).
// D (16x16 f32) is the resident color accumulator: M=pixel, N=channel.
// ---------------------------------------------------------------------------
__global__ __launch_bounds__(256) void gs_rasterize(
    const float4* __restrict__ srecs, const float* __restrict__ bg,
    float* __restrict__ out) {
  __shared__ float4 buf[2][QUADS];       // 24 KB double buffer
  __shared__ float  outb[8][16][16];     // 8 KB D-matrix unpack staging

  const int tid  = threadIdx.x;
  const int wave = tid >> 5;
  const int lane = tid & 31;
  const int pixl = lane & 15;
  const int half = lane >> 4;
  const int pix  = blockIdx.x * 128 + wave * 16 + pixl;   // 512 blocks * 128 px
  const float pfx = (float)(pix & (GW - 1));
  const float pfy = (float)(pix >> 8);

  // --- async stage chunk c into buffer sel (3 x b128 per thread) ---
#define GS_ISSUE(c_, sel_)                                                     \
  do {                                                                         \
    uint32_t lbase_ = (uint32_t)(uintptr_t)(&buf[(sel_)][0]);                  \
    uint32_t goff_  = (uint32_t)((c_) * QUADS * 16) + (uint32_t)tid * 16u;     \
    uint32_t loff_  = lbase_ + (uint32_t)tid * 16u;                            \
    asm volatile("global_load_async_to_lds_b128 %0, %1, %2 offset:0"           \
                 :: "v"(loff_), "v"(goff_), "s"(srecs) : "memory");            \
    asm volatile("global_load_async_to_lds_b128 %0, %1, %2 offset:0"           \
                 :: "v"(loff_ + 4096u), "v"(goff_ + 4096u), "s"(srecs)         \
                 : "memory");                                                  \
    asm volatile("global_load_async_to_lds_b128 %0, %1, %2 offset:0"           \
                 :: "v"(loff_ + 8192u), "v"(goff_ + 8192u), "s"(srecs)         \
                 : "memory");                                                  \
  } while (0)

  GS_ISSUE(0, 0);
  GS_ISSUE(1, 1);

  v8f acc = {0.f, 0.f, 0.f, 0.f, 0.f, 0.f, 0.f, 0.f};
  float T = 1.0f;

  for (int c = 0; c < NCHUNK; ++c) {
    if (c + 1 < NCHUNK) {
      asm volatile("s_wait_asynccnt 0x3" ::: "memory");   // chunk c landed
    } else {
      asm volatile("s_wait_asynccnt 0x0" ::: "memory");
    }
    __syncthreads();

    const float4* cb = &buf[c & 1][0];
    for (int s = 0; s < NG_CHUNK / 4; ++s) {
      const int gA = s * 4 + half * 2;          // this half's first gaussian
      float4 r0 = cb[gA*3+0], r1 = cb[gA*3+1], r2 = cb[gA*3+2];
      float4 q0 = cb[gA*3+3], q1 = cb[gA*3+4], q2 = cb[gA*3+5];

      float dx = r0.x - pfx, dy = r0.y - pfy;
      float pw = -0.5f*(r0.z*dx*dx + r1.x*dy*dy) - r0.w*dx*dy;
      float aA = fminf(0.99f, r1.y * expf(pw));
      aA = (pw > 0.0f || aA < (1.0f/255.0f)) ? 0.0f : aA;

      dx = q0.x - pfx; dy = q0.y - pfy;
      pw = -0.5f*(q0.z*dx*dx + q1.x*dy*dy) - q0.w*dx*dy;
      float aB = fminf(0.99f, q1.y * expf(pw));
      aB = (pw > 0.0f || aB < (1.0f/255.0f)) ? 0.0f : aB;

      float m     = (1.0f - aA) * (1.0f - aB);     // this half's (1-a) product
      float other = __shfl_xor(m, 16);             // other half's product
      float pre   = half ? other : 1.0f;           // exclusive prefix for K2,K3
      float wA = T * pre * aA;
      float wB = T * pre * (1.0f - aA) * aB;
      T *= m * other;

      float bA = (pixl == 0) ? r1.z : (pixl == 1) ? r1.w : (pixl == 2) ? r2.x : 0.0f;
      float bB = (pixl == 0) ? q1.z : (pixl == 1) ? q1.w : (pixl == 2) ? q2.x : 0.0f;

      v2f A = { wA, wB };    // A 16x4 f32: lanes0-15 K0,K1; lanes16-31 K2,K3
      v2f B = { bA, bB };    // B 4x16 f32: lanes0-15 rows K0,K1; 16-31 K2,K3
      acc = __builtin_amdgcn_wmma_f32_16x16x4_f32(
          /*neg_a=*/false, A, /*neg_b=*/false, B,
          /*c_mod=*/(short)0, acc, /*reuse_a=*/false, /*reuse_b=*/false);
    }
    __syncthreads();
    if (c + 2 < NCHUNK) GS_ISSUE(c + 2, c & 1);
  }

  // Unpack D: lane(0-15) holds N=lane, M=r; lane(16-31) holds N=lane-16, M=8+r
#pragma unroll
  for (int r = 0; r < 8; ++r) outb[wave][r + half*8][pixl] = acc[r];
  __syncthreads();

  if (half == 0) {
    float b0 = bg[0], b1 = bg[1], b2 = bg[2];
    out[0*GW*GH + pix] = outb[wave][pixl][0] + T * b0;
    out[1*GW*GH + pix] = outb[wave][pixl][1] + T * b1;
    out[2*GW*GH + pix] = outb[wave][pixl][2] + T * b2;
  }
#undef GS_ISSUE
}

// ---------------------------------------------------------------------------
extern "C" void kernel_launch(void* const* d_in, const int* in_sizes, int n_in,
                              void* d_out, int out_size, void* d_ws, size_t ws_size,
                              hipStream_t stream) {
  (void)in_sizes; (void)n_in; (void)out_size; (void)ws_size;
  const float* means   = (const float*)d_in[0];
  const float* colors  = (const float*)d_in[1];
  const float* opac    = (const float*)d_in[2];
  const float* scales  = (const float*)d_in[3];
  const float* rots    = (const float*)d_in[4];
  const float* bg      = (const float*)d_in[5];
  const float* viewm   = (const float*)d_in[6];
  const float* projm   = (const float*)d_in[7];
  float* out = (float*)d_out;

  char* ws = (char*)d_ws;
  float*  depths = (float*)ws;                                   // 32 KB
  float4* recsU  = (float4*)(ws + 32768);                        // 384 KB
  float4* recsS  = (float4*)(ws + 32768 + (size_t)GP * 48);      // 384 KB

  gs_preprocess<<<GP/256, 256, 0, stream>>>(means, colors, opac, scales, rots,
                                            viewm, projm, depths, recsU);
  gs_sort_gather<<<1, 1024, 0, stream>>>(depths, recsU, recsS);
  gs_rasterize<<<(GW*GH)/128, 256, 0, stream>>>(recsS, bg, out);
}